// ParticlePickingLoss_86603720557297
// MI455X (gfx1250) — compile-verified
//
#include <hip/hip_runtime.h>
#include <hip/hip_bf16.h>
#include <math.h>

typedef __attribute__((ext_vector_type(2))) float v2f;
typedef __attribute__((ext_vector_type(8))) float v8f;

#define NMAT 128   // P
#define NT   96    // T
#define NB   16    // B

__device__ __forceinline__ float softplusf(float x) {
    // numerically stable softplus: max(x,0) + log1p(exp(-|x|))
    return fmaxf(x, 0.f) + log1pf(expf(-fabsf(x)));
}

// ---------------------------------------------------------------------------
// Kernel 1: cost[b][m][n] = sqrt(max(0, |t_m|^2 - 2 t.p + |p_n|^2))
//                          + softplus(x_n) - x_n * y_m
// Gram term (-2 t.p) via V_WMMA_F32_16X16X4_F32: one wave per 16x16 tile.
// A (16x4): row m holds (-2*tx, -2*ty, 0, 0).  B (4x16): col n holds (px,py,0,0).
// ---------------------------------------------------------------------------
__global__ void cost_kernel(const float* __restrict__ tc,    // [B,T,2]
                            const float* __restrict__ pred,  // [B,P,3]
                            float* __restrict__ cost)        // [B,P,P]
{
    const int tile = blockIdx.x;          // 0..63
    const int b    = blockIdx.y;          // 0..15
    const int tm   = tile >> 3;           // row-tile
    const int tn   = tile & 7;            // col-tile
    const int lane = threadIdx.x;         // wave32
    const int half = lane >> 4;
    const int l16  = lane & 15;

    const int n_g = tn * 16 + l16;        // this lane's pred column (fixed)
    const float x  = pred[(b * NMAT + n_g) * 3 + 0];
    const float px = pred[(b * NMAT + n_g) * 3 + 1];
    const float py = pred[(b * NMAT + n_g) * 3 + 2];

    v2f a = {0.f, 0.f};                   // A frag: lanes 16-31 carry K=2,3 (zero)
    v2f bf = {0.f, 0.f};                  // B frag: lanes 16-31 carry K=2,3 rows (zero)
    if (half == 0) {
        const int m_g = tm * 16 + l16;
        float tx = 0.f, ty = 0.f;
        if (m_g < NT) {
            tx = tc[(b * NT + m_g) * 2 + 0];
            ty = tc[(b * NT + m_g) * 2 + 1];
        }
        a[0] = -2.0f * tx;                // bake the -2 factor into A
        a[1] = -2.0f * ty;
        bf[0] = px;
        bf[1] = py;
    }

    v8f c = {};
    // D = A x B (+0): accumulator holds -2 * (tx*px + ty*py) per element
    c = __builtin_amdgcn_wmma_f32_16x16x4_f32(
            /*neg_a=*/false, a, /*neg_b=*/false, bf,
            /*c_mod=*/(short)0, c, /*reuse_a=*/false, /*reuse_b=*/false);

    const float sp    = softplusf(x);
    const float pnorm = px * px + py * py;

    #pragma unroll
    for (int vv = 0; vv < 8; ++vv) {
        const int m   = half * 8 + vv;    // C/D layout: lane<16 -> M=vgpr, else M=vgpr+8
        const int m_g = tm * 16 + m;
        float tx = 0.f, ty = 0.f, y = 0.f;
        if (m_g < NT) {
            tx = tc[(b * NT + m_g) * 2 + 0];
            ty = tc[(b * NT + m_g) * 2 + 1];
            y  = 1.f;
        }
        const float d2    = tx * tx + ty * ty + pnorm + c[vv];
        const float coord = sqrtf(fmaxf(d2, 0.f));
        cost[((size_t)b * NMAT + m_g) * NMAT + n_g] = coord + sp - x * y;
    }
}

// ---------------------------------------------------------------------------
// Kernel 2: Jonker-Volgenant Hungarian. One wave32 block per batch.
// Cost matrix staged into LDS with GLOBAL_LOAD_ASYNC_TO_LDS_B128 (ASYNCcnt),
// then potentials/arrays in LDS; 128-wide min/argmin per augmenting step
// via 5-step wave32 shfl_xor reduction.
// ---------------------------------------------------------------------------
__global__ void hungarian_kernel(const float* __restrict__ cost,  // [B,P,P]
                                 int* __restrict__ col_out)       // [B,P]
{
    const int b    = blockIdx.x;
    const int lane = threadIdx.x;         // 0..31
    extern __shared__ float smem[];
    float* cs   = smem;                   // 128*128 floats
    float* u    = cs + NMAT * NMAT;       // 129 (pad to 132)
    float* v    = u + 132;
    float* minv = v + 132;
    int*   p    = (int*)(minv + 132);
    int*   way  = p + 132;
    int*   used = way + 132;

    const float* cb = cost + (size_t)b * NMAT * NMAT;

    // ---- async global -> LDS stage of the 64KB cost slice -----------------
    // Generic shared pointers carry the LDS byte offset in their low 32 bits
    // (per the aperture spec), giving the VDST operand.
    {
        const unsigned int lds_base = (unsigned int)(unsigned long long)cs;
        const unsigned long long gbase = (unsigned long long)cb;
        for (int k = lane * 16; k < NMAT * NMAT * 4; k += 32 * 16) {
            unsigned int       la = lds_base + (unsigned int)k;
            unsigned long long ga = gbase + (unsigned long long)k;
            asm volatile("global_load_async_to_lds_b128 %0, %1, off"
                         :: "v"(la), "v"(ga) : "memory");
        }
        asm volatile("s_wait_asynccnt 0" ::: "memory");
    }

    for (int k = lane; k <= NMAT; k += 32) { u[k] = 0.f; v[k] = 0.f; p[k] = 0; way[k] = 0; }
    __syncthreads();

    const float INF = __builtin_huge_valf();

    for (int i = 1; i <= NMAT; ++i) {
        if (lane == 0) p[0] = i;
        for (int k = lane; k <= NMAT; k += 32) { minv[k] = INF; used[k] = 0; }
        __syncthreads();

        int j0 = 0;
        while (true) {
            if (lane == 0) used[j0] = 1;
            __syncthreads();

            const int   i0 = p[j0];
            const float u0 = u[i0];

            float best  = INF;
            int   bestj = NMAT + 1;
            for (int j = 1 + lane; j <= NMAT; j += 32) {   // lane owns these j's
                if (!used[j]) {
                    const float cur = cs[(i0 - 1) * NMAT + (j - 1)] - u0 - v[j];
                    float mv = minv[j];
                    if (cur < mv) { mv = cur; minv[j] = cur; way[j] = j0; }
                    // ascending j within lane: strict < keeps first-minimum tiebreak
                    if (mv < best || (mv == best && j < bestj)) { best = mv; bestj = j; }
                }
            }
            // wave32 min+argmin reduction (tie -> smallest j, matching np.argmin)
            #pragma unroll
            for (int off = 16; off > 0; off >>= 1) {
                const float ob = __shfl_xor(best, off, 32);
                const int   oj = __shfl_xor(bestj, off, 32);
                if (ob < best || (ob == best && oj < bestj)) { best = ob; bestj = oj; }
            }
            const float delta = best;
            const int   j1    = bestj;
            __syncthreads();  // minv/way owner differs from updater below

            for (int k = lane; k <= NMAT; k += 32) {
                if (used[k]) { u[p[k]] += delta; v[k] -= delta; }  // p[k] distinct over used k
                else         { minv[k] -= delta; }
            }
            __syncthreads();

            j0 = j1;
            if (p[j0] == 0) break;
        }
        // augment along alternating path (serial, lane 0)
        if (lane == 0) {
            int jj = j0;
            while (jj) { const int jn = way[jj]; p[jj] = p[jn]; jj = jn; }
        }
        __syncthreads();
    }

    for (int j = 1 + lane; j <= NMAT; j += 32)
        col_out[b * NMAT + (p[j] - 1)] = j - 1;
}

// ---------------------------------------------------------------------------
// Kernel 3: matched loss -> scalar.  2048 (b,i) terms, one 256-thread block.
// ---------------------------------------------------------------------------
__global__ void loss_kernel(const float* __restrict__ tc,
                            const float* __restrict__ pred,
                            const int* __restrict__ col,
                            float* __restrict__ out)
{
    __shared__ float red[256];
    const int tid = threadIdx.x;
    float acc = 0.f;
    for (int idx = tid; idx < NB * NMAT; idx += 256) {
        const int b = idx / NMAT;
        const int i = idx % NMAT;
        const int c = col[b * NMAT + i];
        const float* mp = pred + ((size_t)b * NMAT + c) * 3;
        const float x = mp[0], p1 = mp[1], p2 = mp[2];
        float tx = 0.f, ty = 0.f, y = 0.f;
        if (i < NT) {
            tx = tc[(b * NT + i) * 2 + 0];
            ty = tc[(b * NT + i) * 2 + 1];
            y  = 1.f;
        }
        const float dx = p1 - tx, dy = p2 - ty;
        acc += (dx * dx + dy * dy) * (1.0f / 256.0f)       // coord mean over (128,2)
             + (softplusf(x) - x * y) * (1.0f / 128.0f);   // cls mean over 128
    }
    red[tid] = acc;
    __syncthreads();
    for (int s = 128; s > 0; s >>= 1) {
        if (tid < s) red[tid] += red[tid + s];
        __syncthreads();
    }
    if (tid == 0) out[0] = red[0] * (1.0f / (float)NB);    // mean over batch
}

extern "C" void kernel_launch(void* const* d_in, const int* in_sizes, int n_in,
                              void* d_out, int out_size, void* d_ws, size_t ws_size,
                              hipStream_t stream) {
    const float* tc   = (const float*)d_in[0];  // [16,96,2]
    const float* pred = (const float*)d_in[1];  // [16,128,3]
    float* out = (float*)d_out;                 // scalar f32

    float* cost_ws = (float*)d_ws;                               // B*P*P floats = 1 MB
    int*   col_ws  = (int*)((char*)d_ws + (size_t)NB * NMAT * NMAT * sizeof(float));

    // 1) cost matrix: 64 tiles x 16 batches, one wave per 16x16 WMMA tile
    cost_kernel<<<dim3(64, NB), 32, 0, stream>>>(tc, pred, cost_ws);

    // 2) Hungarian: one wave32 block per batch, cost staged in 68 KB dynamic LDS
    const size_t lds = (size_t)NMAT * NMAT * sizeof(float) + 6 * 132 * sizeof(float);
    hungarian_kernel<<<NB, 32, lds, stream>>>(cost_ws, col_ws);

    // 3) final scalar loss
    loss_kernel<<<1, 256, 0, stream>>>(tc, pred, col_ws, out);
}